// GeoLayer_35888746726011
// MI455X (gfx1250) — compile-verified
//
#include <hip/hip_runtime.h>
#include <hip/hip_bf16.h>

#define N_NODES   50000
#define N_EDGES   800000
#define E_TOTAL   (N_EDGES + N_NODES)   // with self loops = 850000
#define IN_CH     256
#define HEADS     8
#define OUT_CH    32
#define HC        (HEADS * OUT_CH)      // 256
#define NEG_SLOPE 0.2f

typedef __attribute__((ext_vector_type(2))) float v2f;
typedef __attribute__((ext_vector_type(8))) float v8f;

// ---------------------------------------------------------------------------
// monotonic float <-> uint encoding for atomicMax on floats
// ---------------------------------------------------------------------------
__device__ __forceinline__ unsigned fenc(float f) {
    unsigned u = __float_as_uint(f);
    return (u & 0x80000000u) ? ~u : (u | 0x80000000u);
}
__device__ __forceinline__ float fdec(unsigned u) {
    unsigned v = (u & 0x80000000u) ? (u & 0x7FFFFFFFu) : ~u;
    return __uint_as_float(v);
}
#define NEG_INF_KEY 0x007FFFFFu   // fenc(-inf)

// ---------------------------------------------------------------------------
// Kernel 0a: out[n, k] = bias[k]
// ---------------------------------------------------------------------------
__global__ void k_init_out(float* __restrict__ out, const float* __restrict__ bias) {
    int idx = blockIdx.x * blockDim.x + threadIdx.x;
    if (idx < N_NODES * HC) out[idx] = bias[idx & (HC - 1)];
}

// ---------------------------------------------------------------------------
// Kernel 0b: mkey = enc(-inf), denom = 0   (N_NODES*HEADS entries)
// ---------------------------------------------------------------------------
__global__ void k_init_md(unsigned* __restrict__ mkey, float* __restrict__ denom) {
    int idx = blockIdx.x * blockDim.x + threadIdx.x;
    if (idx < N_NODES * HEADS) { mkey[idx] = NEG_INF_KEY; denom[idx] = 0.0f; }
}

// ---------------------------------------------------------------------------
// Kernel 1: xw = x @ weight via V_WMMA_F32_16X16X4_F32.
// Grid: 3125 m-tiles. Block: 512 threads = 16 waves, wave w owns n-tile w.
// A 16x4 layout: lane m in [0,16): v0=K0,v1=K1 ; lanes 16-31: v0=K2,v1=K3.
// B 4x16 layout mirrored across N. C/D: VGPR r -> row r (lanes<16) / r+8.
// ---------------------------------------------------------------------------
__global__ __launch_bounds__(512) void k_gemm_wmma(const float* __restrict__ x,
                                                   const float* __restrict__ w,
                                                   float* __restrict__ xw) {
    const int mtile = blockIdx.x;             // 0..3124  (50000/16)
    const int wave  = threadIdx.x >> 5;       // 0..15  -> n-tile
    const int lane  = threadIdx.x & 31;
    const int row0  = mtile * 16;
    const int col0  = wave * 16;
    const int m     = lane & 15;
    const int kk    = (lane >> 4) << 1;       // 0 or 2

    const float* __restrict__ xrow = x + (size_t)(row0 + m) * IN_CH;
    v8f acc = {};
    #pragma unroll 4
    for (int k = 0; k < IN_CH; k += 4) {
        v2f a, b;
        a.x = xrow[k + kk];
        a.y = xrow[k + kk + 1];
        b.x = w[(size_t)(k + kk)     * HC + col0 + m];
        b.y = w[(size_t)(k + kk + 1) * HC + col0 + m];
        acc = __builtin_amdgcn_wmma_f32_16x16x4_f32(
            /*neg_a=*/false, a, /*neg_b=*/false, b,
            /*c_mod=*/(short)0, acc, /*reuse_a=*/false, /*reuse_b=*/false);
    }
    const int orow = row0 + ((lane >> 4) << 3);   // +0 or +8
    const int ocol = col0 + m;
    #pragma unroll
    for (int r = 0; r < 8; ++r)
        xw[(size_t)(orow + r) * HC + ocol] = acc[r];
}

// ---------------------------------------------------------------------------
// Kernel 2: per-node attention scores.
// Block = 256 threads = 8 waves = 8 heads, one block per node.
// s_dst[n,h] = dot(xw[n,h,:], att[0,h,0:32])   (pairs with destination index)
// s_src[n,h] = dot(xw[n,h,:], att[0,h,32:64])  (pairs with source index)
// ---------------------------------------------------------------------------
__global__ __launch_bounds__(256) void k_scores(const float* __restrict__ xw,
                                                const float* __restrict__ att,
                                                float* __restrict__ s_dst,
                                                float* __restrict__ s_src) {
    const int n = blockIdx.x;
    const int h = threadIdx.x >> 5;
    const int c = threadIdx.x & 31;
    const float v  = xw[(size_t)n * HC + h * OUT_CH + c];
    float pi = v * att[h * (2 * OUT_CH) + c];
    float pj = v * att[h * (2 * OUT_CH) + OUT_CH + c];
    #pragma unroll
    for (int off = 16; off > 0; off >>= 1) {
        pi += __shfl_down(pi, off, 32);
        pj += __shfl_down(pj, off, 32);
    }
    if (c == 0) {
        s_dst[n * HEADS + h] = pi;
        s_src[n * HEADS + h] = pj;
    }
}

// ---------------------------------------------------------------------------
// Kernel 3: alpha[e,h] = leakyrelu(s_dst[i]+s_src[j]); atomic max keyed by j.
// One thread per (edge, head).
// ---------------------------------------------------------------------------
__global__ void k_alpha_max(const long long* __restrict__ ei,
                            const float* __restrict__ s_dst,
                            const float* __restrict__ s_src,
                            float* __restrict__ ealpha,
                            unsigned* __restrict__ mkey) {
    const int t = blockIdx.x * blockDim.x + threadIdx.x;
    const int e = t >> 3;
    const int h = t & 7;
    if (e >= E_TOTAL) return;
    int j, i;
    if (e < N_EDGES) { j = (int)ei[e]; i = (int)ei[N_EDGES + e]; }
    else             { j = i = e - N_EDGES; }
    float a = s_dst[i * HEADS + h] + s_src[j * HEADS + h];
    a = (a >= 0.0f) ? a : NEG_SLOPE * a;
    ealpha[(size_t)e * HEADS + h] = a;
    atomicMax(&mkey[j * HEADS + h], fenc(a));
}

// ---------------------------------------------------------------------------
// Kernel 4: ealpha <- exp(alpha - m[j]); denom[j] += ealpha.
// ---------------------------------------------------------------------------
__global__ void k_exp_sum(const long long* __restrict__ ei,
                          float* __restrict__ ealpha,
                          const unsigned* __restrict__ mkey,
                          float* __restrict__ denom) {
    const int t = blockIdx.x * blockDim.x + threadIdx.x;
    const int e = t >> 3;
    const int h = t & 7;
    if (e >= E_TOTAL) return;
    int j;
    if (e < N_EDGES) j = (int)ei[e];
    else             j = e - N_EDGES;
    const float m = fdec(mkey[j * HEADS + h]);
    const float v = expf(ealpha[(size_t)e * HEADS + h] - m);
    ealpha[(size_t)e * HEADS + h] = v;
#if defined(__has_builtin) && __has_builtin(__builtin_amdgcn_global_atomic_fadd_f32)
    unsafeAtomicAdd(&denom[j * HEADS + h], v);
#else
    unsafeAtomicAdd(&denom[j * HEADS + h], v);
#endif
}

// ---------------------------------------------------------------------------
// Kernel 5: out[i, h*32+c] += (ealpha[e,h]/(denom[j,h]+eps)) * xw[j, h*32+c]
// Block = 256 threads per edge; wave = head, so coef is lane-uniform.
// ---------------------------------------------------------------------------
__global__ __launch_bounds__(256) void k_aggregate(const long long* __restrict__ ei,
                                                   const float* __restrict__ xw,
                                                   const float* __restrict__ ealpha,
                                                   const float* __restrict__ denom,
                                                   float* __restrict__ out) {
    const int e = blockIdx.x;
    const int h = threadIdx.x >> 5;
    const int c = threadIdx.x & 31;
    int j, i;
    if (e < N_EDGES) { j = (int)ei[e]; i = (int)ei[N_EDGES + e]; }
    else             { j = i = e - N_EDGES; }
    const float coef = ealpha[(size_t)e * HEADS + h] /
                       (denom[j * HEADS + h] + 1e-16f);
    const float val = coef * xw[(size_t)j * HC + h * OUT_CH + c];
    unsafeAtomicAdd(&out[(size_t)i * HC + h * OUT_CH + c], val);
}

// ---------------------------------------------------------------------------
// Launch
// ---------------------------------------------------------------------------
extern "C" void kernel_launch(void* const* d_in, const int* in_sizes, int n_in,
                              void* d_out, int out_size, void* d_ws, size_t ws_size,
                              hipStream_t stream) {
    const float*     x    = (const float*)d_in[0];      // 50000 x 256
    const float*     wgt  = (const float*)d_in[1];      // 256 x 256
    const float*     att  = (const float*)d_in[2];      // 1 x 8 x 64
    const float*     bias = (const float*)d_in[3];      // 256
    const long long* ei   = (const long long*)d_in[4];  // 2 x 800000 (int64)
    float* out = (float*)d_out;                         // 50000 x 256

    // workspace carve-up (floats)
    float*    xw     = (float*)d_ws;                                // 12,800,000
    float*    s_dst  = xw + (size_t)N_NODES * HC;                   //    400,000
    float*    s_src  = s_dst + (size_t)N_NODES * HEADS;             //    400,000
    unsigned* mkey   = (unsigned*)(s_src + (size_t)N_NODES * HEADS);//    400,000
    float*    denom  = (float*)(mkey + (size_t)N_NODES * HEADS);    //    400,000
    float*    ealpha = denom + (size_t)N_NODES * HEADS;             //  6,800,000
    (void)in_sizes; (void)n_in; (void)out_size; (void)ws_size;

    // init
    k_init_out<<<(N_NODES * HC + 255) / 256, 256, 0, stream>>>(out, bias);
    k_init_md<<<(N_NODES * HEADS + 255) / 256, 256, 0, stream>>>(mkey, denom);

    // 1) xw = x @ weight  (fp32 WMMA)
    k_gemm_wmma<<<N_NODES / 16, 512, 0, stream>>>(x, wgt, xw);

    // 2) per-node scores
    k_scores<<<N_NODES, 256, 0, stream>>>(xw, att, s_dst, s_src);

    // 3) alpha + segment max
    {
        long long tot = (long long)E_TOTAL * HEADS;
        int blocks = (int)((tot + 255) / 256);
        k_alpha_max<<<blocks, 256, 0, stream>>>(ei, s_dst, s_src, ealpha, mkey);
        // 4) exp + segment sum
        k_exp_sum<<<blocks, 256, 0, stream>>>(ei, ealpha, mkey, denom);
    }

    // 5) weighted scatter-add into out
    k_aggregate<<<E_TOTAL, 256, 0, stream>>>(ei, xw, ealpha, denom, out);
}